// CTN_LT_Loss_41566693491289
// MI455X (gfx1250) — compile-verified
//
#include <hip/hip_runtime.h>
#include <stdint.h>

#define L_LEN   30000
#define NV4     (L_LEN / 4)                   // 7500 float4/int4 per row (30000 % 4 == 0)
#define BLK     1024
#define NCH4    ((NV4 + BLK - 1) / BLK)       // 8 vector chunks (last partial: 332)
#define TOPM    50
#define ALPHA_C 0.8f
#define EPS_C   1e-8f
#define CAP_GT  1600                          // > 49*32 = 1568 (provable bound)
#define CAP_EQ  64                            // >= TOPM slots for == T0 ties

// dynamic LDS layout (bytes)
#define OFF_BUF   0          // float buf[30000]      : logits, then bce (in place)
#define OFF_BITS  120000     // unsigned bits[938]    : linear target bitmap
#define OFF_TVAL  123840     // float tval[1024]      : per-thread bce maxima
#define OFF_CAND  127936     // float cand[1664]      : candidates (gt | eq regions)
#define OFF_SRED  134592     // float sred[32]
#define OFF_IRED  134720     // int   ired[32]
#define OFF_SCAL  134848     // int   scal[4] : [0]=cgt [1]=ceq [2]=T0 key
#define SMEM_BYTES 134912

typedef unsigned int v4u __attribute__((ext_vector_type(4)));
typedef int          v4i __attribute__((ext_vector_type(4)));
typedef int          v8i __attribute__((ext_vector_type(8)));

__device__ __forceinline__ unsigned ordkey(float f) {
    unsigned b = __float_as_uint(f);
    return (b >> 31) ? ~b : (b | 0x80000000u);   // monotone total order on f32
}
__device__ __forceinline__ float ordinv(unsigned u) {
    return __uint_as_float((u & 0x80000000u) ? (u ^ 0x80000000u) : ~u);
}
__device__ __forceinline__ float wsum_f(float v) {
    for (int o = 16; o; o >>= 1) v += __shfl_down(v, o, 32);
    return v;
}
__device__ __forceinline__ int wsum_i(int v) {
    for (int o = 16; o; o >>= 1) v += __shfl_down(v, o, 32);
    return v;
}
__device__ __forceinline__ float wmax_f(float v) {
    for (int o = 16; o; o >>= 1) v = fmaxf(v, __shfl_down(v, o, 32));
    return v;
}
// 3-barrier block reductions (1024 threads = exactly 32 waves)
__device__ __forceinline__ float block_sum(float v, float* sred, int lane, int wave) {
    v = wsum_f(v);
    if (lane == 0) sred[wave] = v;
    __syncthreads();
    if (wave == 0) { float x = wsum_f(sred[lane]); if (lane == 0) sred[0] = x; }
    __syncthreads();
    float r = sred[0];
    __syncthreads();
    return r;
}
__device__ __forceinline__ float block_max(float v, float* sred, int lane, int wave) {
    v = wmax_f(v);
    if (lane == 0) sred[wave] = v;
    __syncthreads();
    if (wave == 0) { float x = wmax_f(sred[lane]); if (lane == 0) sred[0] = x; }
    __syncthreads();
    float r = sred[0];
    __syncthreads();
    return r;
}
__device__ __forceinline__ void block_sum_fi(float v, int c, float* sred, int* ired,
                                             int lane, int wave, float* ov, int* oc) {
    v = wsum_f(v); c = wsum_i(c);
    if (lane == 0) { sred[wave] = v; ired[wave] = c; }
    __syncthreads();
    if (wave == 0) {
        float x = wsum_f(sred[lane]); int y = wsum_i(ired[lane]);
        if (lane == 0) { sred[0] = x; ired[0] = y; }
    }
    __syncthreads();
    *ov = sred[0]; *oc = ired[0];
    __syncthreads();
}

__global__ __launch_bounds__(BLK) void ctn_row_kernel(
    const float* __restrict__ logits, const int* __restrict__ targets,
    float* __restrict__ ws, int rows)
{
    extern __shared__ char smem[];
    float*    buf  = (float*)(smem + OFF_BUF);
    float4*   buf4 = (float4*)(smem + OFF_BUF);
    unsigned* bits = (unsigned*)(smem + OFF_BITS);
    float*    tval = (float*)(smem + OFF_TVAL);
    float*    cand = (float*)(smem + OFF_CAND);
    float*    sred = (float*)(smem + OFF_SRED);
    int*      ired = (int*)(smem + OFF_IRED);
    int*      scal = (int*)(smem + OFF_SCAL);

    const int row  = blockIdx.x;
    const int tid  = threadIdx.x;
    const int lane = tid & 31;
    const int wave = tid >> 5;
    const float NEG_INF = -__builtin_inff();

    // ---- 1) TDM: async DMA of the 120KB logits row into LDS ----
    if (tid == 0) {
        unsigned long long ga = (unsigned long long)(uintptr_t)(logits + (size_t)row * L_LEN);
        unsigned lds_off = (unsigned)(uintptr_t)buf;   // addr[31:0] == LDS byte offset
        v4u g0;
        g0.x = 1u;                                          // count=1 (valid user D#)
        g0.y = lds_off;                                     // lds_addr
        g0.z = (unsigned)ga;                                // global_addr[31:0]
        g0.w = (unsigned)((ga >> 32) & 0x1FFFFFFu) | (2u << 30); // addr[56:32] | type=2
        v8i g1;
        g1[0] = 0x00020000;                                 // data_size=2 -> 4 bytes
        g1[1] = (int)((L_LEN & 0xFFFF) << 16);              // tensor_dim0 lo16
        g1[2] = (int)((L_LEN >> 16) | (1 << 16));           // tensor_dim0 hi | tensor_dim1=1 lo
        g1[3] = (int)((L_LEN & 0xFFFF) << 16);              // tensor_dim1 hi(0) | tile_dim0
        g1[4] = 1;                                          // tile_dim1=1, tile_dim2=0
        g1[5] = L_LEN;                                      // tensor_dim0_stride lo32
        g1[6] = 0; g1[7] = 0;
        v4i z4 = {0, 0, 0, 0};
        v8i z8 = {0, 0, 0, 0, 0, 0, 0, 0};
        __builtin_amdgcn_tensor_load_to_lds(g0, g1, z4, z4, z8, 0);
        scal[0] = 0; scal[1] = 0;                           // compaction counters
    }

    // ---- 2) stream targets (B128) -> linear LDS bitmap (overlaps TDM) ----
    {
        const int4* t4 = (const int4*)(targets + (size_t)row * L_LEN);
        for (int c = 0; c < NCH4; ++c) {
            int idx = c * BLK + tid;                        // int4 index
            int4 v = make_int4(0, 0, 0, 0);
            if (idx < NV4) v = t4[idx];
            unsigned nib = (unsigned)(v.x == 1) | ((unsigned)(v.y == 1) << 1) |
                           ((unsigned)(v.z == 1) << 2) | ((unsigned)(v.w == 1) << 3);
            unsigned wbits = nib << ((lane & 7) * 4);
            wbits |= __shfl_xor(wbits, 1, 32);              // OR-combine groups of 8 lanes
            wbits |= __shfl_xor(wbits, 2, 32);
            wbits |= __shfl_xor(wbits, 4, 32);
            if ((lane & 7) == 0 && idx < NV4) bits[idx >> 3] = wbits;
        }
    }
    if (tid == 0) __builtin_amdgcn_s_wait_tensorcnt(0);
    __syncthreads();

    // ---- 3) row max over all logits (B128 LDS reads) ----
    float m = NEG_INF;
    for (int c = 0; c < NCH4; ++c) {
        int idx = c * BLK + tid;
        if (idx < NV4) {
            float4 v = buf4[idx];
            m = fmaxf(m, fmaxf(fmaxf(v.x, v.y), fmaxf(v.z, v.w)));
        }
    }
    float rowmax = block_max(m, sred, lane, wave);

    // ---- 4) s_neg = sum over negatives of exp(x - rowmax) ----
    float sn = 0.f;
    for (int c = 0; c < NCH4; ++c) {
        int idx = c * BLK + tid;
        if (idx < NV4) {
            float4 v = buf4[idx];
            unsigned nib = (bits[idx >> 3] >> ((idx & 7) * 4)) & 0xFu;
            if (!(nib & 1u)) sn += expf(v.x - rowmax);
            if (!(nib & 2u)) sn += expf(v.y - rowmax);
            if (!(nib & 4u)) sn += expf(v.z - rowmax);
            if (!(nib & 8u)) sn += expf(v.w - rowmax);
        }
    }
    float s_neg = block_sum(sn, sred, lane, wave);

    // ---- 5) per_pos for positives + BCE (in place, B128) + per-thread bce max ----
    float ce_loc = 0.f; int np_loc = 0; float mv = NEG_INF;
    for (int c = 0; c < NCH4; ++c) {
        int idx = c * BLK + tid;
        if (idx < NV4) {
            float4 v = buf4[idx];
            unsigned nib = (bits[idx >> 3] >> ((idx & 7) * 4)) & 0xFu;
            float xs[4] = {v.x, v.y, v.z, v.w};
            float bs[4];
#pragma unroll
            for (int k = 0; k < 4; ++k) {
                float x = xs[k];
                bool pos = (nib >> k) & 1u;
                float e = expf(x - rowmax);
                if (pos) { ce_loc += rowmax + logf(e + s_neg) - x; np_loc++; }
                float p = 1.0f / (1.0f + expf(-x));
                float b = pos ? -logf(p + EPS_C) : -logf(1.0f - p + EPS_C);
                bs[k] = b;
                mv = fmaxf(mv, b);
            }
            buf4[idx] = make_float4(bs[0], bs[1], bs[2], bs[3]); // owned by this thread
        }
    }
    tval[tid] = mv;
    float ce_row; int np_row;
    block_sum_fi(ce_loc, np_loc, sred, ired, lane, wave, &ce_row, &np_row);
    // (barriers above also publish tval[] and in-place bce buf[])

    // ---- 6a) wave 0: T0 = 50th-largest per-thread max (radix search, wave-local) ----
    if (wave == 0) {
        unsigned T0 = 0;
        for (int bit = 31; bit >= 0; --bit) {
            unsigned thr = T0 | (1u << bit);
            int c = 0;
            for (int j = lane; j < BLK; j += 32) c += (ordkey(tval[j]) >= thr) ? 1 : 0;
            c = wsum_i(c);
            c = __shfl(c, 0, 32);
            if (c >= TOPM) T0 = thr;
        }
        if (lane == 0) scal[2] = (int)T0;
    }
    __syncthreads();

    // ---- 6b) compaction: all elems > T0 (<=1568 provably) + <=TOPM elems == T0 ----
    {
        unsigned T0 = (unsigned)scal[2];
        for (int c = 0; c < NCH4; ++c) {
            int idx = c * BLK + tid;
            if (idx < NV4) {
                float4 v = buf4[idx];
                float xs[4] = {v.x, v.y, v.z, v.w};
#pragma unroll
                for (int k = 0; k < 4; ++k) {
                    unsigned kk = ordkey(xs[k]);
                    if (kk > T0) {
                        int p = atomicAdd(&scal[0], 1);
                        if (p < CAP_GT) cand[p] = xs[k];
                    } else if (kk == T0) {
                        int e = atomicAdd(&scal[1], 1);
                        if (e < TOPM) cand[CAP_GT + e] = xs[k];
                    }
                }
            }
        }
    }
    __syncthreads();

    // ---- 6c) wave 0: exact 50th key T among candidates, then top-50 sum ----
    if (wave == 0) {
        int cg = scal[0]; if (cg > CAP_GT) cg = CAP_GT;
        int ne = scal[1]; if (ne > TOPM) ne = TOPM;
        int n = cg + ne;
        unsigned T = 0;
        for (int bit = 31; bit >= 0; --bit) {
            unsigned thr = T | (1u << bit);
            int c = 0;
            for (int j = lane; j < n; j += 32) {
                float f = (j < cg) ? cand[j] : cand[CAP_GT + j - cg];
                c += (ordkey(f) >= thr) ? 1 : 0;
            }
            c = wsum_i(c);
            c = __shfl(c, 0, 32);
            if (c >= TOPM) T = thr;
        }
        float s = 0.f; int cgt = 0;
        for (int j = lane; j < n; j += 32) {
            float f = (j < cg) ? cand[j] : cand[CAP_GT + j - cg];
            if (ordkey(f) > T) { s += f; cgt++; }
        }
        s = wsum_f(s); cgt = wsum_i(cgt);
        if (lane == 0) {
            float topsum = s + (float)(TOPM - cgt) * ordinv(T);  // exact tie handling
            ws[row]            = ce_row;
            ws[rows + row]     = (float)np_row;                  // exact integer in f32
            ws[2 * rows + row] = topsum * (1.0f / (float)TOPM);
        }
    }
}

__global__ __launch_bounds__(256) void ctn_finalize(const float* __restrict__ ws,
                                                    float* __restrict__ out, int rows)
{
    __shared__ float s_ce[256]; __shared__ int s_np[256]; __shared__ float s_mb[256];
    int tid = threadIdx.x;
    float ce = 0.f, mb = 0.f; int np = 0;
    for (int i = tid; i < rows; i += 256) {
        ce += ws[i];
        np += (int)ws[rows + i];                 // exact integer accumulation
        mb += ws[2 * rows + i];
    }
    s_ce[tid] = ce; s_np[tid] = np; s_mb[tid] = mb; __syncthreads();
    for (int s = 128; s > 0; s >>= 1) {
        if (tid < s) { s_ce[tid] += s_ce[tid + s]; s_np[tid] += s_np[tid + s]; s_mb[tid] += s_mb[tid + s]; }
        __syncthreads();
    }
    if (tid == 0) {
        float npf  = (float)s_np[0];
        float cev  = (npf > 0.f) ? (s_ce[0] / fmaxf(npf, 1.0f)) : 0.f;
        float mbce = s_mb[0] / (float)rows;
        out[0] = ALPHA_C * cev + (1.0f - ALPHA_C) * mbce;
        out[1] = cev;
        out[2] = mbce;
    }
}

extern "C" void kernel_launch(void* const* d_in, const int* in_sizes, int n_in,
                              void* d_out, int out_size, void* d_ws, size_t ws_size,
                              hipStream_t stream) {
    const float* logits  = (const float*)d_in[0];
    const int*   targets = (const int*)d_in[1];
    float* out = (float*)d_out;
    float* ws  = (float*)d_ws;
    int rows = in_sizes[0] / L_LEN;   // 2048

    (void)hipFuncSetAttribute(reinterpret_cast<const void*>(ctn_row_kernel),
                              hipFuncAttributeMaxDynamicSharedMemorySize, SMEM_BYTES);

    ctn_row_kernel<<<rows, BLK, SMEM_BYTES, stream>>>(logits, targets, ws, rows);
    ctn_finalize<<<1, 256, 0, stream>>>(ws, out, rows);
}